// ECELoss_20066087207578
// MI455X (gfx1250) — compile-verified
//
#include <hip/hip_runtime.h>
#include <math.h>

#define NBINS 10

typedef float v4f __attribute__((ext_vector_type(4)));

// ---------------------------------------------------------------------------
// Zero the 30 double accumulators in workspace (counts | sum_conf | sum_acc).
// ---------------------------------------------------------------------------
__global__ void __launch_bounds__(64)
ece_zero_ws(double* __restrict__ ws) {
    int t = threadIdx.x;
    if (t < 3 * NBINS) ws[t] = 0.0;
}

// ---------------------------------------------------------------------------
// Wave32 max+argmax reduction, first-occurrence tie-break (matches jnp.argmax)
// ---------------------------------------------------------------------------
__device__ __forceinline__ void wave_argmax(float& m, int& mi) {
    #pragma unroll
    for (int off = 16; off > 0; off >>= 1) {
        float om = __shfl_xor(m, off, 32);
        int   oi = __shfl_xor(mi, off, 32);
        if (om > m || (om == m && oi < mi)) { m = om; mi = oi; }
    }
}

__device__ __forceinline__ void bin_row(float m, int mi, int lbl,
                                        float* s_cnt, float* s_conf, float* s_acc) {
    float conf = expf(m);                         // (1/C, 1] for log-softmax rows
    int bin = (int)ceilf(conf * (float)NBINS) - 1;
    bin = bin < 0 ? 0 : (bin > NBINS - 1 ? NBINS - 1 : bin);
    float acc = (mi == lbl) ? 1.0f : 0.0f;
    atomicAdd(&s_cnt[bin],  1.0f);                // ds_add_f32, non-returning
    atomicAdd(&s_conf[bin], conf);
    atomicAdd(&s_acc[bin],  acc);
}

// ---------------------------------------------------------------------------
// Main pass: one row per wave32 per iteration.
//   lanes 0..(C/4-1) each load one float4 of the row (coalesced 400B burst,
//   non-temporal: 800MB streams once, don't thrash the 192MB L2).
//   Common case (C%4==0, C<=128): software-pipelined — the next row's
//   global_load_b128 is issued BEFORE reducing the current row, so HBM
//   latency overlaps the shuffle reduction + LDS atomics.
// ---------------------------------------------------------------------------
__global__ void __launch_bounds__(256)
ece_partial(const float* __restrict__ logits,
            const int*   __restrict__ labels,
            double*      __restrict__ ws,
            int n_rows, int n_cols) {
    __shared__ float s_cnt[NBINS];
    __shared__ float s_conf[NBINS];
    __shared__ float s_acc[NBINS];

    const int tid = threadIdx.x;
    if (tid < NBINS) { s_cnt[tid] = 0.f; s_conf[tid] = 0.f; s_acc[tid] = 0.f; }
    __syncthreads();

    const int lane          = tid & 31;
    const int waveInBlock   = tid >> 5;
    const int wavesPerBlock = blockDim.x >> 5;
    const long long stride  = (long long)gridDim.x * wavesPerBlock;
    long long row0 = (long long)blockIdx.x * wavesPerBlock + waveInBlock;

    const int c4    = n_cols >> 2;   // full float4 chunks per row (25 for C=100)
    const int ctail = n_cols & 3;    // leftover scalars (0 for C=100)

    if (ctail == 0 && c4 <= 32) {
        // ---- specialized pipelined path: one b128 load per row per wave ----
        const bool active = lane < c4;
        long long row = row0;
        v4f cur = {0.f, 0.f, 0.f, 0.f};
        if (row < n_rows && active) {
            cur = __builtin_nontemporal_load(
                      (const v4f*)(logits + row * (long long)n_cols) + lane);
        }
        for (; row < n_rows; row += stride) {
            // issue next row's load first (overlaps with reduction below)
            long long nrow = row + stride;
            v4f nxt = {0.f, 0.f, 0.f, 0.f};
            if (nrow < n_rows && active) {
                nxt = __builtin_nontemporal_load(
                          (const v4f*)(logits + nrow * (long long)n_cols) + lane);
            }
            int lbl = labels[row];   // wave-uniform -> scalar load, overlaps too

            float m  = -3.402823466e+38f;
            int   mi = 0x7fffffff;
            if (active) {
                int b = lane << 2;
                m = cur.x; mi = b;
                if (cur.y > m) { m = cur.y; mi = b + 1; }
                if (cur.z > m) { m = cur.z; mi = b + 2; }
                if (cur.w > m) { m = cur.w; mi = b + 3; }
            }
            wave_argmax(m, mi);
            if (lane == 0) bin_row(m, mi, lbl, s_cnt, s_conf, s_acc);
            cur = nxt;
        }
    } else {
        // ---- generic fallback (any C) ----
        for (long long row = row0; row < n_rows; row += stride) {
            const float* rowp = logits + row * (long long)n_cols;
            long long nrow = row + stride;
            if (nrow < n_rows && lane < c4) {   // gfx1250 global_prefetch_b8
                __builtin_prefetch(logits + nrow * (long long)n_cols + 4 * lane, 0, 1);
            }
            float m  = -3.402823466e+38f;
            int   mi = 0x7fffffff;
            for (int c = lane; c < c4; c += 32) {
                v4f v = __builtin_nontemporal_load((const v4f*)rowp + c);
                int b = c << 2;
                if (v.x > m) { m = v.x; mi = b;     }
                if (v.y > m) { m = v.y; mi = b + 1; }
                if (v.z > m) { m = v.z; mi = b + 2; }
                if (v.w > m) { m = v.w; mi = b + 3; }
            }
            if (lane < ctail) {
                int b = (c4 << 2) + lane;
                float v = rowp[b];
                if (v > m) { m = v; mi = b; }
            }
            wave_argmax(m, mi);
            if (lane == 0) bin_row(m, mi, labels[row], s_cnt, s_conf, s_acc);
        }
    }

    __syncthreads();
    if (tid < NBINS) {
        atomicAdd(&ws[tid],             (double)s_cnt[tid]);   // global_atomic_add_f64
        atomicAdd(&ws[NBINS + tid],     (double)s_conf[tid]);
        atomicAdd(&ws[2 * NBINS + tid], (double)s_acc[tid]);
    }
}

// ---------------------------------------------------------------------------
// Finalize: ECE = sum over non-empty bins of |avg_conf - avg_acc| * (cnt / n)
// ---------------------------------------------------------------------------
__global__ void __launch_bounds__(32)
ece_final(const double* __restrict__ ws, float* __restrict__ out, int n_rows) {
    if (threadIdx.x == 0 && blockIdx.x == 0) {
        double ece = 0.0;
        double n   = (double)n_rows;
        #pragma unroll
        for (int i = 0; i < NBINS; ++i) {
            double cnt = ws[i];
            if (cnt > 0.0) {
                double gap = fabs(ws[NBINS + i] / cnt - ws[2 * NBINS + i] / cnt);
                ece += gap * (cnt / n);
            }
        }
        out[0] = (float)ece;
    }
}

// ---------------------------------------------------------------------------
extern "C" void kernel_launch(void* const* d_in, const int* in_sizes, int n_in,
                              void* d_out, int out_size, void* d_ws, size_t ws_size,
                              hipStream_t stream) {
    const float* logits = (const float*)d_in[0];
    const int*   labels = (const int*)d_in[1];
    float*  out = (float*)d_out;
    double* ws  = (double*)d_ws;

    const int n_rows = in_sizes[1];                 // N = 2,000,000
    const int n_cols = in_sizes[0] / n_rows;        // C = 100

    ece_zero_ws<<<1, 64, 0, stream>>>(ws);

    const int block = 256;
    const int wavesPerBlock = block / 32;
    long long wavesNeeded = n_rows;
    int blocks = (int)((wavesNeeded + wavesPerBlock - 1) / wavesPerBlock);
    if (blocks > 4096) blocks = 4096;
    if (blocks < 1)    blocks = 1;

    ece_partial<<<blocks, block, 0, stream>>>(logits, labels, ws, n_rows, n_cols);
    ece_final<<<1, 32, 0, stream>>>(ws, out, n_rows);
}